// GuidedAttentionLoss_16930761081222
// MI455X (gfx1250) — compile-verified
//
#include <hip/hip_runtime.h>
#include <hip/hip_bf16.h>

// GuidedAttentionLoss on MI455X (gfx1250, wave32).
//
// Memory-bound scalar reduction: 134 MB of f32 attention probs -> 1 float.
// Strategy:
//  * 1 block per (batch b, 8 target rows): 256 threads, each thread owns one
//    float4 (16B) slot per row-pair -> global_load_b128, fully coalesced.
//  * Skip loads for invalid rows (t >= olen, wave-uniform branch) and for
//    lanes whose whole float4 is past ilen -> cuts HBM traffic ~2-4x.
//  * Non-temporal loads (read-once stream, don't pollute 192MB L2).
//  * Wave reduction via V_WMMA_F32_16X16X4_F32 (A = accs in documented 16x4
//    32-bit layout, B = ones): D[m,n] = acc(m)+acc(m+16); sum 8 D VGPRs and
//    one shfl_xor(16) finishes the wave. Matrix pipe co-executes with VALU.
//  * Deterministic: block partials -> d_ws, fixed-order final reduction
//    kernel (no float atomics). den = sum_b olen*ilen computed exactly.

typedef float v2f __attribute__((ext_vector_type(2)));
typedef float v4f __attribute__((ext_vector_type(4)));
typedef float v8f __attribute__((ext_vector_type(8)));

#define T_DIM 2048
#define S_DIM 512
#define ROWS_PER_BLOCK 8
#define THREADS 256
#define SIGMA 0.4f

__global__ __launch_bounds__(THREADS)
void ga_partial_kernel(const int* __restrict__ source_len,
                       const int* __restrict__ target_len,
                       const float* __restrict__ attn,
                       float* __restrict__ partials)
{
    const int b   = blockIdx.y;
    const int t0  = blockIdx.x * ROWS_PER_BLOCK;
    const int tid = threadIdx.x;

    const int ilen = source_len[b];
    const int olen = target_len[b];
    const float inv_il = 1.0f / (float)ilen;
    const float inv_ol = 1.0f / (float)olen;
    const float kExp   = -1.0f / (2.0f * SIGMA * SIGMA);   // -3.125

    const int s4   = tid & 127;   // which float4 within the S=512 row
    const int s0   = s4 << 2;     // first source index of this float4
    const int rsub = tid >> 7;    // 0/1: row within the pair handled per iter

    float acc = 0.0f;

    #pragma unroll
    for (int i = 0; i < ROWS_PER_BLOCK / 2; ++i) {
        const int t = t0 + i * 2 + rsub;
        // t < olen is uniform across each wave (128 lanes share a row ->
        // every wave32 shares t); skipping saves real HBM bandwidth.
        if (t < olen && s0 < ilen) {
            const float* p = attn + (((size_t)b * T_DIM + t) * S_DIM + s0);
            v4f v = __builtin_nontemporal_load((const v4f*)p);
            const float xo = (float)t * inv_ol;
            #pragma unroll
            for (int j = 0; j < 4; ++j) {
                const int s = s0 + j;
                if (s < ilen) {
                    const float d = (float)s * inv_il - xo;
                    const float w = 1.0f - __expf(d * d * kExp);
                    acc += w * v[j];
                }
            }
        }
    }

    // ---- wave32 reduction via V_WMMA_F32_16X16X4_F32 ----
    // 32-bit A 16x4 layout: lanes 0-15 = M rows, VGPR0->K0, VGPR1->K1;
    // lanes 16-31 = same M rows, VGPR0->K2, VGPR1->K3.
    // a = {acc, 0}  =>  A[m,0] = acc(lane m), A[m,2] = acc(lane m+16).
    // B = ones      =>  D[m,n] = acc(m) + acc(m+16)  (layout-independent).
    v2f a;    a[0] = acc;  a[1] = 0.0f;
    v2f ones; ones[0] = 1.0f; ones[1] = 1.0f;
    v8f c = {};
    v8f dm = __builtin_amdgcn_wmma_f32_16x16x4_f32(
        /*neg_a=*/false, a, /*neg_b=*/false, ones,
        /*c_mod=*/(short)0, c, /*reuse_a=*/false, /*reuse_b=*/false);

    // D layout: lane n<16 VGPR v = D[v, n] (rows 0-7); lane>=16 -> rows 8-15.
    float half_sum = dm[0] + dm[1] + dm[2] + dm[3]
                   + dm[4] + dm[5] + dm[6] + dm[7];
    float wave_sum = half_sum + __shfl_xor(half_sum, 16, 32);

    __shared__ float lds[THREADS / 32];
    const int wave = tid >> 5;
    if ((tid & 31) == 0) lds[wave] = wave_sum;
    __syncthreads();

    if (tid == 0) {
        float s = 0.0f;
        #pragma unroll
        for (int wv = 0; wv < THREADS / 32; ++wv) s += lds[wv];
        partials[(size_t)blockIdx.y * gridDim.x + blockIdx.x] = s;
    }
}

// Single-block, fixed-order (deterministic) final reduction + exact den.
__global__ __launch_bounds__(256)
void ga_final_kernel(const int* __restrict__ source_len,
                     const int* __restrict__ target_len,
                     const float* __restrict__ partials,
                     int n_partials,
                     float* __restrict__ out,
                     int B)
{
    __shared__ float lds[256];
    float s = 0.0f;
    for (int i = threadIdx.x; i < n_partials; i += 256)
        s += partials[i];                      // fixed stride order
    lds[threadIdx.x] = s;
    __syncthreads();
    #pragma unroll
    for (int stride = 128; stride > 0; stride >>= 1) {
        if ((int)threadIdx.x < stride)
            lds[threadIdx.x] += lds[threadIdx.x + stride];
        __syncthreads();
    }
    if (threadIdx.x == 0) {
        long long den = 0;
        for (int bb = 0; bb < B; ++bb)
            den += (long long)source_len[bb] * (long long)target_len[bb];
        out[0] = lds[0] / (float)den;
    }
}

extern "C" void kernel_launch(void* const* d_in, const int* in_sizes, int n_in,
                              void* d_out, int out_size, void* d_ws, size_t ws_size,
                              hipStream_t stream) {
    const int*   source_len = (const int*)d_in[0];
    const int*   target_len = (const int*)d_in[1];
    const float* attn       = (const float*)d_in[2];
    float*       partials   = (float*)d_ws;   // 8192 floats = 32 KB
    float*       out        = (float*)d_out;

    const int B = in_sizes[0];                       // 32
    const int blocks_x = T_DIM / ROWS_PER_BLOCK;     // 256
    dim3 grid(blocks_x, B);

    ga_partial_kernel<<<grid, THREADS, 0, stream>>>(source_len, target_len,
                                                    attn, partials);
    ga_final_kernel<<<1, 256, 0, stream>>>(source_len, target_len, partials,
                                           blocks_x * B, out, B);
}